// Decoder_1967095022178
// MI455X (gfx1250) — compile-verified
//
#include <hip/hip_runtime.h>
#include <math.h>

// ---- problem constants (match reference) ----
#define BSZ    32
#define TLEN   20
#define TM     19        // TLEN - 1
#define HDIM   256
#define EDIM   512
#define ADIM   256
#define VDIM   10000
#define DDIM   256
#define PDIM   1024      // GH*GW
#define INFLAT (PDIM*DDIM)   // 262144
#define RELN   16384
#define XDIM   1024      // E + D + H

typedef __attribute__((ext_vector_type(2))) float v2f;
typedef __attribute__((ext_vector_type(8))) float v8f;

// ---------------- CDNA5 fp32 WMMA: D(16x16) += A(16x4) * B(4x16) -------------
__device__ __forceinline__ v8f wmma4(v2f a, v2f b, v8f c) {
  return __builtin_amdgcn_wmma_f32_16x16x4_f32(false, a, false, b, (short)0, c,
                                               false, false);
}

// Fragment layout (wave32):
//  A 16x4: lane L (L<16) holds row M=L, K = k+0 (v0), k+1 (v1);
//          lane L (>=16) holds row M=L-16, K = k+2 (v0), k+3 (v1).
//  B 4x16: lane L holds col N=L&15; K = k + 2*(L>>4) + {0,1}.
//  C/D 16x16: VGPR g, lane L -> M = g + 8*(L>>4), N = L&15.

// ======== dual-M-tile WMMA GEMM: C(Mx N) = A @ B + bias, 32 rows / wave ======
// Each wave owns one 16-wide N tile and BOTH 16-row M tiles of a 32-row band:
// the B (weight) fragment is loaded once and feeds two WMMAs -> half the
// weight-fetch instructions for the HBM/L2-bound streams.
__global__ void wmma_gemm_bias_m2(const float* __restrict__ A,
                                  const float* __restrict__ B,
                                  const float* __restrict__ bias,
                                  float* __restrict__ C, int M, int K, int N) {
  const int lane = threadIdx.x & 31;
  const int wv   = threadIdx.x >> 5;
  const int tn   = (blockIdx.x * (blockDim.x >> 5) + wv) << 4;
  if (tn >= N) return;                 // wave-uniform; EXEC stays all-1 for WMMA
  const int tm   = blockIdx.y << 5;    // 32-row band
  const int hi   = lane >> 4;
  const int l15  = lane & 15;
  const float* Ap0 = A + (size_t)(tm + l15) * K + (hi << 1);
  const float* Ap1 = Ap0 + (size_t)16 * K;
  const float* Bp  = B + (size_t)(hi << 1) * N + tn + l15;
  v8f acc0 = {}, acc1 = {};
  #pragma unroll 4
  for (int k = 0; k < K; k += 4) {
    v2f b, a0, a1;
    b.x  = Bp[(size_t)k * N];
    b.y  = Bp[(size_t)(k + 1) * N];
    a0.x = Ap0[k];  a0.y = Ap0[k + 1];
    a1.x = Ap1[k];  a1.y = Ap1[k + 1];
    acc0 = wmma4(a0, b, acc0);
    acc1 = wmma4(a1, b, acc1);
  }
  const float bb = bias[tn + l15];
  #pragma unroll
  for (int g = 0; g < 8; ++g) {
    C[(size_t)(tm + (hi << 3) + g) * N + tn + l15]      = acc0[g] + bb;
    C[(size_t)(tm + 16 + (hi << 3) + g) * N + tn + l15] = acc1[g] + bb;
  }
}

// ======== K-split dual-M-tile GEMM for the 537 MB weight streams (M==32) =====
__global__ void wmma_gemm_ksplit_m2(const float* __restrict__ A,
                                    const float* __restrict__ B,
                                    float* __restrict__ part,
                                    int K, int N, int kchunk) {
  const int lane = threadIdx.x & 31;
  const int wv   = threadIdx.x >> 5;
  const int tn   = (blockIdx.x * (blockDim.x >> 5) + wv) << 4;
  if (tn >= N) return;
  const int s    = blockIdx.z;
  const int k0   = s * kchunk;
  int k1 = k0 + kchunk; if (k1 > K) k1 = K;
  const int hi   = lane >> 4;
  const int l15  = lane & 15;
  const float* Ap0 = A + (size_t)l15 * K + (hi << 1);
  const float* Ap1 = Ap0 + (size_t)16 * K;
  const float* Bp  = B + (size_t)(hi << 1) * N + tn + l15;
  v8f acc0 = {}, acc1 = {};
  #pragma unroll 4
  for (int k = k0; k < k1; k += 4) {
    v2f b, a0, a1;
    b.x  = Bp[(size_t)k * N];
    b.y  = Bp[(size_t)(k + 1) * N];
    a0.x = Ap0[k];  a0.y = Ap0[k + 1];
    a1.x = Ap1[k];  a1.y = Ap1[k + 1];
    acc0 = wmma4(a0, b, acc0);
    acc1 = wmma4(a1, b, acc1);
  }
  float* Pp = part + (size_t)s * BSZ * N;
  #pragma unroll
  for (int g = 0; g < 8; ++g) {
    Pp[(size_t)((hi << 3) + g) * N + tn + l15]      = acc0[g];
    Pp[(size_t)(16 + (hi << 3) + g) * N + tn + l15] = acc1[g];
  }
}

__global__ void reduce_partials(const float* __restrict__ part,
                                const float* __restrict__ bias,
                                float* __restrict__ outp, int S, int N, int total) {
  int idx = blockIdx.x * blockDim.x + threadIdx.x;
  if (idx >= total) return;
  float s = 0.f;
  for (int i = 0; i < S; ++i) s += part[(size_t)i * total + idx];
  outp[idx] = s + bias[idx % N];
}

// ======== all 5 gate GEMMs in one launch (grid.z selects gate) ===============
struct GateArgs { const float* W[5]; const float* b[5]; };

__global__ void wmma_gates(const float* __restrict__ x, GateArgs ga,
                           float* __restrict__ gates) {
  const int g5   = blockIdx.z;
  const float* B    = ga.W[g5];
  const float* bias = ga.b[g5];
  float* C = gates + (size_t)g5 * BSZ * HDIM;
  const int lane = threadIdx.x & 31;
  const int wv   = threadIdx.x >> 5;
  const int tn   = (blockIdx.x * (blockDim.x >> 5) + wv) << 4;
  if (tn >= HDIM) return;
  const int hi   = lane >> 4;
  const int l15  = lane & 15;
  const float* Ap0 = x + (size_t)l15 * XDIM + (hi << 1);
  const float* Ap1 = Ap0 + (size_t)16 * XDIM;
  const float* Bp  = B + (size_t)(hi << 1) * HDIM + tn + l15;
  v8f acc0 = {}, acc1 = {};
  #pragma unroll 4
  for (int k = 0; k < XDIM; k += 4) {
    v2f b, a0, a1;
    b.x  = Bp[(size_t)k * HDIM];
    b.y  = Bp[(size_t)(k + 1) * HDIM];
    a0.x = Ap0[k];  a0.y = Ap0[k + 1];
    a1.x = Ap1[k];  a1.y = Ap1[k + 1];
    acc0 = wmma4(a0, b, acc0);
    acc1 = wmma4(a1, b, acc1);
  }
  const float bb = bias[tn + l15];
  #pragma unroll
  for (int g = 0; g < 8; ++g) {
    C[(size_t)((hi << 3) + g) * HDIM + tn + l15]      = acc0[g] + bb;
    C[(size_t)(16 + (hi << 3) + g) * HDIM + tn + l15] = acc1[g] + bb;
  }
}

// ======== FC GEMM fused with masking + output placement (M==32) ==============
// out[(b*TM + t)*N + n] = (h @ W_fc + b_fc) * (sent_len[b] > t)
__global__ void wmma_fc_pred(const float* __restrict__ A,
                             const float* __restrict__ B,
                             const float* __restrict__ bias,
                             float* __restrict__ out,
                             const int* __restrict__ sent_len,
                             int t, int K, int N) {
  const int lane = threadIdx.x & 31;
  const int wv   = threadIdx.x >> 5;
  const int tn   = (blockIdx.x * (blockDim.x >> 5) + wv) << 4;
  if (tn >= N) return;
  const int hi   = lane >> 4;
  const int l15  = lane & 15;
  const float* Ap0 = A + (size_t)l15 * K + (hi << 1);
  const float* Ap1 = Ap0 + (size_t)16 * K;
  const float* Bp  = B + (size_t)(hi << 1) * N + tn + l15;
  v8f acc0 = {}, acc1 = {};
  #pragma unroll 4
  for (int k = 0; k < K; k += 4) {
    v2f b, a0, a1;
    b.x  = Bp[(size_t)k * N];
    b.y  = Bp[(size_t)(k + 1) * N];
    a0.x = Ap0[k];  a0.y = Ap0[k + 1];
    a1.x = Ap1[k];  a1.y = Ap1[k + 1];
    acc0 = wmma4(a0, b, acc0);
    acc1 = wmma4(a1, b, acc1);
  }
  const float bb = bias[tn + l15];
  #pragma unroll
  for (int g = 0; g < 8; ++g) {
    int b0 = (hi << 3) + g;            // batch rows 0..15
    int b1 = b0 + 16;                  // batch rows 16..31
    float m0 = (sent_len[b0] > t) ? 1.f : 0.f;
    float m1 = (sent_len[b1] > t) ? 1.f : 0.f;
    out[((size_t)b0 * TM + t) * N + tn + l15] = (acc0[g] + bb) * m0;
    out[((size_t)b1 * TM + t) * N + tn + l15] = (acc1[g] + bb) * m1;
  }
}

// ---------------- stable descending argsort of lens (B=32) ------------------
__global__ void compute_order(const int* __restrict__ lens, int* __restrict__ order,
                              int* __restrict__ sent_len, float* __restrict__ order_f) {
  int i = threadIdx.x;
  if (i >= BSZ) return;
  int li = lens[i];
  int rank = 0;
  for (int j = 0; j < BSZ; ++j) {
    int lj = lens[j];
    if (lj > li || (lj == li && j < i)) rank++;
  }
  order[rank] = i;
  sent_len[rank] = li - 1;
  order_f[rank] = (float)i;
}

// ---------------- batch gathers ----------------------------------------------
__global__ void gather_rows(const float* __restrict__ src, const int* __restrict__ order,
                            float* __restrict__ dst, int rowlen, long total) {
  long idx = (long)blockIdx.x * blockDim.x + threadIdx.x;
  if (idx >= total) return;
  int b = (int)(idx / rowlen);
  long r = idx % rowlen;
  dst[idx] = src[(size_t)order[b] * rowlen + r];
}

__global__ void gather_words(const int* __restrict__ captions, const int* __restrict__ order,
                             const float* __restrict__ emb, float* __restrict__ words) {
  long idx = (long)blockIdx.x * blockDim.x + threadIdx.x;   // B*TLEN*EDIM
  if (idx >= (long)BSZ * TLEN * EDIM) return;
  int e = (int)(idx % EDIM);
  int t = (int)((idx / EDIM) % TLEN);
  int b = (int)(idx / ((long)EDIM * TLEN));
  int tok = captions[order[b] * TLEN + t];
  words[idx] = emb[(size_t)tok * EDIM + e];
}

// ---------------- attention score: e = relu(att1 + att2) . W_full + b --------
__global__ void att_e_kernel(const float* __restrict__ att1, const float* __restrict__ att2,
                             const float* __restrict__ W_full, const float* __restrict__ b_full,
                             float* __restrict__ e) {
  int wave = blockIdx.x * (blockDim.x >> 5) + (threadIdx.x >> 5);   // b*P + p
  int lane = threadIdx.x & 31;
  if (wave >= BSZ * PDIM) return;
  int b = wave >> 10;
  const float* a1 = att1 + (size_t)wave * ADIM;
  const float* a2 = att2 + (size_t)b * ADIM;
  float s = 0.f;
  #pragma unroll 2
  for (int k = lane; k < ADIM; k += 32) {
    float v = a1[k] + a2[k];
    v = v > 0.f ? v : 0.f;
    s += v * W_full[k];
  }
  #pragma unroll
  for (int off = 16; off; off >>= 1) s += __shfl_down(s, off, 32);
  if (lane == 0) e[wave] = s + b_full[0];
}

// ---------------- softmax over P, plus masked alpha output -------------------
__global__ void softmax_alpha(const float* __restrict__ e, float* __restrict__ alpha,
                              float* __restrict__ alpha_out, const int* __restrict__ sent_len,
                              int t) {
  __shared__ float red[256];
  int b = blockIdx.x;
  const float* eb = e + (size_t)b * PDIM;
  float m = -1e30f;
  for (int p = threadIdx.x; p < PDIM; p += 256) m = fmaxf(m, eb[p]);
  red[threadIdx.x] = m;
  __syncthreads();
  for (int s = 128; s; s >>= 1) {
    if (threadIdx.x < s) red[threadIdx.x] = fmaxf(red[threadIdx.x], red[threadIdx.x + s]);
    __syncthreads();
  }
  m = red[0];
  __syncthreads();
  float sum = 0.f;
  for (int p = threadIdx.x; p < PDIM; p += 256) sum += __expf(eb[p] - m);
  red[threadIdx.x] = sum;
  __syncthreads();
  for (int s = 128; s; s >>= 1) {
    if (threadIdx.x < s) red[threadIdx.x] += red[threadIdx.x + s];
    __syncthreads();
  }
  float inv = 1.f / red[0];
  float msk = (sent_len[b] > t) ? 1.f : 0.f;
  for (int p = threadIdx.x; p < PDIM; p += 256) {
    float a = __expf(eb[p] - m) * inv;
    alpha[(size_t)b * PDIM + p] = a;
    alpha_out[((size_t)b * TM + t) * PDIM + p] = a * msk;
  }
}

// ---------------- attention pooling: awf[b][d] = sum_p alpha * info ----------
__global__ void awf_kernel(const float* __restrict__ alpha, const float* __restrict__ info_r,
                           float* __restrict__ awf) {
  int b = blockIdx.x, d = threadIdx.x;    // 32 blocks x 256 threads
  const float* ib = info_r + (size_t)b * PDIM * DDIM + d;
  const float* ab = alpha + (size_t)b * PDIM;
  float s = 0.f;
  for (int p = 0; p < PDIM; ++p) s += ab[p] * ib[(size_t)p * DDIM];
  awf[b * DDIM + d] = s;
}

// ---------------- x = [word_t, awf, h] ---------------------------------------
__global__ void build_x(const float* __restrict__ words, const float* __restrict__ awf,
                        const float* __restrict__ h, float* __restrict__ x, int t) {
  int idx = blockIdx.x * blockDim.x + threadIdx.x;   // B * XDIM
  if (idx >= BSZ * XDIM) return;
  int b = idx >> 10, j = idx & 1023;
  float v;
  if (j < EDIM)             v = words[((size_t)b * TLEN + t) * EDIM + j];
  else if (j < EDIM + DDIM) v = awf[b * DDIM + (j - EDIM)];
  else                      v = h[b * HDIM + (j - EDIM - DDIM)];
  x[idx] = v;
}

__device__ __forceinline__ float sigm(float v) { return 1.f / (1.f + __expf(-v)); }

// ---------------- LSTM elementwise + mlp-input concat ------------------------
__global__ void lstm_cell(const float* __restrict__ gi, const float* __restrict__ gf,
                          const float* __restrict__ go, const float* __restrict__ g1,
                          const float* __restrict__ g2, const float* __restrict__ c,
                          const float* __restrict__ Cc, float* __restrict__ c_new,
                          float* __restrict__ C_new, float* __restrict__ o_act,
                          float* __restrict__ mlp_in) {
  int idx = blockIdx.x * blockDim.x + threadIdx.x;   // B * H
  if (idx >= BSZ * HDIM) return;
  int b = idx >> 8, hh = idx & 255;
  float iv = sigm(gi[idx]), fv = sigm(gf[idx]), ov = sigm(go[idx]);
  float G1 = tanhf(g1[idx]), G2 = tanhf(g2[idx]);
  float nc = fv * c[idx] + iv * G1;
  float nC = fv * Cc[idx] + iv * G2;
  c_new[idx] = nc; C_new[idx] = nC; o_act[idx] = ov;
  mlp_in[(size_t)b * (2 * HDIM) + hh] = nc;
  mlp_in[(size_t)b * (2 * HDIM) + HDIM + hh] = nC;
}

__global__ void hnew_kernel(const float* __restrict__ o_act, const float* __restrict__ mlp_out,
                            float* __restrict__ h_new) {
  int idx = blockIdx.x * blockDim.x + threadIdx.x;
  if (idx >= BSZ * HDIM) return;
  h_new[idx] = o_act[idx] * tanhf(mlp_out[idx]);
}

// =============================================================================
extern "C" void kernel_launch(void* const* d_in, const int* in_sizes, int n_in,
                              void* d_out, int out_size, void* d_ws, size_t ws_size,
                              hipStream_t stream) {
  const float* info     = (const float*)d_in[0];
  const float* relation = (const float*)d_in[1];
  const int*   captions = (const int*)d_in[2];
  const int*   lens     = (const int*)d_in[3];
  const float* emb      = (const float*)d_in[4];
  const float* W_enc    = (const float*)d_in[5];
  const float* b_enc    = (const float*)d_in[6];
  const float* W_dec    = (const float*)d_in[7];
  const float* b_dec    = (const float*)d_in[8];
  const float* W_full   = (const float*)d_in[9];
  const float* b_full   = (const float*)d_in[10];
  GateArgs ga;
  ga.W[0] = (const float*)d_in[11]; ga.b[0] = (const float*)d_in[12];
  ga.W[1] = (const float*)d_in[13]; ga.b[1] = (const float*)d_in[14];
  ga.W[2] = (const float*)d_in[15]; ga.b[2] = (const float*)d_in[16];
  ga.W[3] = (const float*)d_in[17]; ga.b[3] = (const float*)d_in[18];
  ga.W[4] = (const float*)d_in[19]; ga.b[4] = (const float*)d_in[20];
  const float* W_mlp    = (const float*)d_in[21];
  const float* b_mlp    = (const float*)d_in[22];
  // d_in[23], d_in[24] = W_beta, b_beta: unused by reference
  const float* W_fc     = (const float*)d_in[25];
  const float* b_fc     = (const float*)d_in[26];
  const float* W_ih     = (const float*)d_in[27];
  const float* b_ih     = (const float*)d_in[28];
  const float* W_ic     = (const float*)d_in[29];
  const float* b_ic     = (const float*)d_in[30];
  const float* W_iC     = (const float*)d_in[31];
  const float* b_iC     = (const float*)d_in[32];

  float* out = (float*)d_out;
  const size_t PRED_OFF  = 0;                               // 32*19*10000
  const size_t ALPHA_OFF = (size_t)BSZ * TM * VDIM;         // 6,080,000
  const size_t ORDER_OFF = ALPHA_OFF + (size_t)BSZ * TM * PDIM;

  // ---- workspace layout (ints first, then 256B-aligned floats) ----
  char* wsb = (char*)d_ws;
  int* order_i = (int*)wsb;
  int* sentlen = order_i + BSZ;
  float* f = (float*)(wsb + 256);
  size_t off = 0;
  #define WALLOC(name, n) float* name = f + off; off += (size_t)(n)
  WALLOC(info_r,  (size_t)BSZ * INFLAT);       // 8.4M  (also "flat" view)
  WALLOC(rel_r,   (size_t)BSZ * RELN);
  WALLOC(words_r, (size_t)BSZ * TLEN * EDIM);
  WALLOC(att1,    (size_t)BSZ * PDIM * ADIM);  // 8.4M
  WALLOC(partial, (size_t)128 * BSZ * HDIM);   // K-split partials
  WALLOC(h0, BSZ * HDIM); WALLOC(h1, BSZ * HDIM);
  WALLOC(c0, BSZ * HDIM); WALLOC(c1, BSZ * HDIM);
  WALLOC(C0, BSZ * HDIM); WALLOC(C1, BSZ * HDIM);
  WALLOC(att2,  BSZ * ADIM);
  WALLOC(evec,  BSZ * PDIM);
  WALLOC(alpha, BSZ * PDIM);
  WALLOC(awf,   BSZ * DDIM);
  WALLOC(xbuf,  BSZ * XDIM);
  WALLOC(gates, 5 * BSZ * HDIM);
  WALLOC(mlp_in,  BSZ * 2 * HDIM);
  WALLOC(mlp_out, BSZ * HDIM);
  WALLOC(o_act,   BSZ * HDIM);
  #undef WALLOC
  (void)ws_size; (void)n_in; (void)in_sizes; (void)out_size;

  // ---- 1) order / sorted lengths ----
  compute_order<<<1, 32, 0, stream>>>(lens, order_i, sentlen, out + ORDER_OFF);

  // ---- 2) batch reorder gathers ----
  {
    long tot = (long)BSZ * INFLAT;
    gather_rows<<<(unsigned)((tot + 255) / 256), 256, 0, stream>>>(info, order_i, info_r, INFLAT, tot);
    tot = (long)BSZ * RELN;
    gather_rows<<<(unsigned)((tot + 255) / 256), 256, 0, stream>>>(relation, order_i, rel_r, RELN, tot);
    tot = (long)BSZ * TLEN * EDIM;
    gather_words<<<(unsigned)((tot + 255) / 256), 256, 0, stream>>>(captions, order_i, emb, words_r);
  }

  // ---- 3) big streaming GEMMs, K-split + dual M tiles (537 MB of weights) ----
  // h = flat @ W_ih + b_ih   (K=262144): S=128 chunks of 2048
  wmma_gemm_ksplit_m2<<<dim3(4, 1, 128), 128, 0, stream>>>(info_r, W_ih, partial,
                                                           INFLAT, HDIM, 2048);
  reduce_partials<<<(BSZ * HDIM + 255) / 256, 256, 0, stream>>>(partial, b_ih, h0,
                                                                128, HDIM, BSZ * HDIM);
  // c = flat @ W_ic + b_ic
  wmma_gemm_ksplit_m2<<<dim3(4, 1, 128), 128, 0, stream>>>(info_r, W_ic, partial,
                                                           INFLAT, HDIM, 2048);
  reduce_partials<<<(BSZ * HDIM + 255) / 256, 256, 0, stream>>>(partial, b_ic, c0,
                                                                128, HDIM, BSZ * HDIM);
  // C = rel @ W_iC + b_iC   (K=16384): S=16 chunks of 1024
  wmma_gemm_ksplit_m2<<<dim3(4, 1, 16), 128, 0, stream>>>(rel_r, W_iC, partial,
                                                          RELN, HDIM, 1024);
  reduce_partials<<<(BSZ * HDIM + 255) / 256, 256, 0, stream>>>(partial, b_iC, C0,
                                                                16, HDIM, BSZ * HDIM);

  // ---- 4) att1 = info_r @ W_enc + b_enc   (M = B*P = 32768 = 1024 x 32) ----
  wmma_gemm_bias_m2<<<dim3(ADIM / 64, (BSZ * PDIM) / 32), 128, 0, stream>>>(
      info_r, W_enc, b_enc, att1, BSZ * PDIM, DDIM, ADIM);

  // ---- 5) recurrent loop (19 steps); L2-resident working set ----
  float *hc = h0, *hn = h1, *cc = c0, *cn = c1, *Cc = C0, *Cn = C1;
  for (int t = 0; t < TM; ++t) {
    // att2 = h @ W_dec + b_dec
    wmma_gemm_bias_m2<<<dim3(ADIM / 64, 1), 128, 0, stream>>>(
        hc, W_dec, b_dec, att2, BSZ, HDIM, ADIM);
    // e = relu(att1 + att2) . W_full + b_full
    att_e_kernel<<<(BSZ * PDIM) / 8, 256, 0, stream>>>(att1, att2, W_full, b_full, evec);
    // alpha = softmax(e), masked copy to output
    softmax_alpha<<<BSZ, 256, 0, stream>>>(evec, alpha, out + ALPHA_OFF, sentlen, t);
    // awf = alpha . info_r
    awf_kernel<<<BSZ, DDIM, 0, stream>>>(alpha, info_r, awf);
    // x = [word_t, awf, h]
    build_x<<<(BSZ * XDIM) / 256, 256, 0, stream>>>(words_r, awf, hc, xbuf, t);
    // 5 gate GEMMs in one launch (M=32, K=1024, N=256)
    wmma_gates<<<dim3(HDIM / 64, 1, 5), 128, 0, stream>>>(xbuf, ga, gates);
    // elementwise LSTM update + concat for MLP
    lstm_cell<<<(BSZ * HDIM) / 256, 256, 0, stream>>>(
        gates, gates + BSZ * HDIM, gates + 2 * BSZ * HDIM, gates + 3 * BSZ * HDIM,
        gates + 4 * BSZ * HDIM, cc, Cc, cn, Cn, o_act, mlp_in);
    // mlp_out = [c,C] @ W_mlp + b_mlp
    wmma_gemm_bias_m2<<<dim3(HDIM / 64, 1), 128, 0, stream>>>(
        mlp_in, W_mlp, b_mlp, mlp_out, BSZ, 2 * HDIM, HDIM);
    // h_new = o * tanh(mlp_out)
    hnew_kernel<<<(BSZ * HDIM) / 256, 256, 0, stream>>>(o_act, mlp_out, hn);
    // predictions: (h_new @ W_fc + b_fc) * mask, written in place (N=10000=625*16)
    wmma_fc_pred<<<dim3((625 + 3) / 4, 1), 128, 0, stream>>>(
        hn, W_fc, b_fc, out + PRED_OFF, sentlen, t, HDIM, VDIM);
    // swap state
    float* tp;
    tp = hc; hc = hn; hn = tp;
    tp = cc; cc = cn; cn = tp;
    tp = Cc; Cc = Cn; Cn = tp;
  }
}